// HeteroGNN_87986700026404
// MI455X (gfx1250) — compile-verified
//
#include <hip/hip_runtime.h>

// ---------------------------------------------------------------------------
// HeteroGNN (2 branches: jaccard + blast) on MI455X (gfx1250, wave32).
// Dense GEMMs: V_WMMA_F32_16X16X4_F32 (exact fp32). One wave owns a 16xM
// output strip: A fragment reused across all column tiles (8x less A traffic),
// W staged in LDS (320KB/WGP) via one coalesced cooperative load per block.
// Edge aggregation: float4 gathers + global_atomic_add_f32 scatters (the
// true bottleneck: ~4.6 GB of HBM traffic -> ~0.2 ms at 23.3 TB/s).
// ---------------------------------------------------------------------------

typedef __attribute__((ext_vector_type(2))) float v2f;
typedef __attribute__((ext_vector_type(8))) float v8f;

#define NN 100000
#define EE 500000

// ---------------------------------------------------------------- utilities

__global__ void fill_kernel(float* __restrict__ p, float v, long long n) {
  long long i = (long long)blockIdx.x * blockDim.x + threadIdx.x;
  if (i < n) p[i] = v;
}

__global__ void count_kernel(const int* __restrict__ col, float* __restrict__ cnt, int e) {
  int i = blockIdx.x * blockDim.x + threadIdx.x;
  if (i < e) atomicAdd(&cnt[col[i]], 1.0f);
}

__global__ void rsqrt_kernel(float* __restrict__ p, int n) {
  int i = blockIdx.x * blockDim.x + threadIdx.x;
  if (i < n) p[i] = rsqrtf(p[i]);   // deg >= 1 (self loop), always valid
}

__global__ void relu_kernel(float* __restrict__ p, long long n) {
  long long i = (long long)blockIdx.x * blockDim.x + threadIdx.x;
  if (i < n) p[i] = fmaxf(p[i], 0.0f);
}

// ------------------------------------------------------ WMMA fp32 strip GEMM
// C[r, coff+c] (ldc stride) = (acc ? C_old : 0) + A[r,:128] @ W[:128, c]
//                             + bias[c], optionally ReLU'd.
// One wave per 16-row strip, all MT columns. K stepped by 4 via
// v_wmma_f32_16x16x4_f32; W held in LDS (row-major, conflict-free reads).

template <int MT>   // MT = output width: 64 or 128 (multiple of 16)
__global__ void gemm_strip_wmma(const float* __restrict__ A, const float* __restrict__ W,
                                const float* __restrict__ bias, float* __restrict__ C,
                                int nrows, int ldc, int coff, int acc, int do_relu) {
  constexpr int K  = 128;
  constexpr int CT = MT / 16;
  __shared__ float lw[K * MT];                       // 64KB (MT=128) / 32KB (MT=64)

  // cooperative, fully-coalesced staging of W into LDS (before any divergence)
  for (int i = threadIdx.x; i < K * MT; i += blockDim.x) lw[i] = W[i];
  __syncthreads();

  const int wave  = threadIdx.x >> 5;
  const int lane  = threadIdx.x & 31;
  const int strip = blockIdx.x * (blockDim.x >> 5) + wave;
  if (strip * 16 >= nrows) return;                   // wave-uniform: EXEC all-1s below

  const int lm   = lane & 15;
  const int half = lane >> 4;
  const int arow = strip * 16 + lm;                  // A row handled by this lane

  v8f accv[CT];
  if (acc) {
#pragma unroll
    for (int ct = 0; ct < CT; ++ct)
#pragma unroll
      for (int r = 0; r < 8; ++r) {
        const int grow = strip * 16 + r + half * 8;
        accv[ct][r] = C[(long long)grow * ldc + coff + ct * 16 + lm];
      }
  } else {
#pragma unroll
    for (int ct = 0; ct < CT; ++ct)
#pragma unroll
      for (int r = 0; r < 8; ++r) accv[ct][r] = 0.0f;
  }

  const float* __restrict__ arp = A + (long long)arow * K;
#pragma unroll 4
  for (int kb = 0; kb < K; kb += 4) {
    const int ka = kb + half * 2;                    // lanes 16-31 carry K=2,3
    v2f a;
    a.x = arp[ka];
    a.y = arp[ka + 1];
#pragma unroll
    for (int ct = 0; ct < CT; ++ct) {
      v2f b;
      b.x = lw[ka * MT + ct * 16 + lm];              // consecutive lanes -> consecutive banks
      b.y = lw[(ka + 1) * MT + ct * 16 + lm];
      accv[ct] = __builtin_amdgcn_wmma_f32_16x16x4_f32(
          /*neg_a=*/false, a, /*neg_b=*/false, b,
          /*c_mod=*/(short)0, accv[ct], /*reuse_a=*/false, /*reuse_b=*/false);
    }
  }

#pragma unroll
  for (int ct = 0; ct < CT; ++ct) {
    const float bv = bias ? bias[ct * 16 + lm] : 0.0f;
#pragma unroll
    for (int r = 0; r < 8; ++r) {
      const int grow = strip * 16 + r + half * 8;
      float v = accv[ct][r] + bv;
      if (do_relu) v = fmaxf(v, 0.0f);
      C[(long long)grow * ldc + coff + ct * 16 + lm] = v;
    }
  }
}

// -------------------------------------------------- GCN self-loop + bias init
// out[i, coff+d] = t[i,d] * dinv[i]^2 + bias[d]   (full overwrite of region)

__global__ void selfloop_init_kernel(const float* __restrict__ t, const float* __restrict__ dinv,
                                     const float* __restrict__ bias, float* __restrict__ out,
                                     int n, int D, int ldc, int coff) {
  long long idx = (long long)blockIdx.x * blockDim.x + threadIdx.x;
  if (idx >= (long long)n * D) return;
  const int i = (int)(idx / D);
  const int d = (int)(idx - (long long)i * D);
  const float di = dinv[i];
  out[(long long)i * ldc + coff + d] = t[idx] * di * di + bias[d];
}

// ------------------------------------------- GCN normalized edge scatter-add
// out[col, coff+d] += t[row, d] * dinv[row] * dinv[col]   (float4 per thread)

__global__ void gcn_scatter_kernel(const float* __restrict__ t, const int* __restrict__ row,
                                   const int* __restrict__ col, const float* __restrict__ dinv,
                                   float* __restrict__ out, int e, int D, int ldc, int coff) {
  const int q = D >> 2;
  long long idx = (long long)blockIdx.x * blockDim.x + threadIdx.x;
  if (idx >= (long long)e * q) return;
  const int ei = (int)(idx / q);
  const int c4 = (int)(idx - (long long)ei * q) << 2;
  const int r  = row[ei];
  const int cd = col[ei];
  const float nrm = dinv[r] * dinv[cd];
  const float4 v = *(const float4*)(t + (long long)r * D + c4);
  float* dst = out + (long long)cd * ldc + coff + c4;
  atomicAdd(dst + 0, v.x * nrm);
  atomicAdd(dst + 1, v.y * nrm);
  atomicAdd(dst + 2, v.z * nrm);
  atomicAdd(dst + 3, v.w * nrm);
}

// ----------------------------------------------- SAGE unnormalized scatter-add
__global__ void sum_scatter_kernel(const float* __restrict__ t, const int* __restrict__ row,
                                   const int* __restrict__ col, float* __restrict__ out,
                                   int e, int D) {
  const int q = D >> 2;
  long long idx = (long long)blockIdx.x * blockDim.x + threadIdx.x;
  if (idx >= (long long)e * q) return;
  const int ei = (int)(idx / q);
  const int c4 = (int)(idx - (long long)ei * q) << 2;
  const int r  = row[ei];
  const int cd = col[ei];
  const float4 v = *(const float4*)(t + (long long)r * D + c4);
  float* dst = out + (long long)cd * D + c4;
  atomicAdd(dst + 0, v.x);
  atomicAdd(dst + 1, v.y);
  atomicAdd(dst + 2, v.z);
  atomicAdd(dst + 3, v.w);
}

// ------------------------------------------------- SAGE mean + bias accumulate
// out[i, coff+d] += s[i,d] / max(cnt[i],1) + bias[d]
__global__ void mean_add_kernel(const float* __restrict__ s, const float* __restrict__ cnt,
                                const float* __restrict__ bias, float* __restrict__ out,
                                int n, int D, int ldc, int coff) {
  long long idx = (long long)blockIdx.x * blockDim.x + threadIdx.x;
  if (idx >= (long long)n * D) return;
  const int i = (int)(idx / D);
  const int d = (int)(idx - (long long)i * D);
  const float c = fmaxf(cnt[i], 1.0f);
  out[(long long)i * ldc + coff + d] += s[idx] / c + bias[d];
}

// ---------------------------------------------------------------------------

static inline int gblk(long long n, int t) { return (int)((n + t - 1) / t); }

extern "C" void kernel_launch(void* const* d_in, const int* in_sizes, int n_in,
                              void* d_out, int out_size, void* d_ws, size_t ws_size,
                              hipStream_t stream) {
  const int N = NN, E = EE;
  const int T = 256;                                   // 8 wave32 per block

  float* ws      = (float*)d_ws;
  float* dinv_ll = ws;
  float* dinv_pp = ws + N;
  float* cnt_lp  = ws + 2 * (size_t)N;
  float* h_l     = ws + 3 * (size_t)N;
  float* h_p     = h_l + (size_t)N * 128;
  float* tmp     = h_p + (size_t)N * 128;
  float* sbuf    = tmp + (size_t)N * 128;

  float* out_lnc  = (float*)d_out;                     // [N,128]: j cols 0-63, b cols 64-127
  float* out_prot = (float*)d_out + (size_t)N * 128;   // [N,128]

  const int strips     = N / 16;                       // 6250, exact
  const int gemmBlocks = gblk(strips, 8);              // 8 strips (waves) per block

  for (int g = 0; g < 2; ++g) {
    const int base = g * 19;
    const float* x_l   = (const float*)d_in[base + 0];
    const float* x_p   = (const float*)d_in[base + 1];
    const int*   ei_lp = (const int*)d_in[base + 2];
    const int*   ei_ll = (const int*)d_in[base + 3];
    const int*   ei_pp = (const int*)d_in[base + 4];
    const float* s1_wl = (const float*)d_in[base + 5];
    const float* s1_bl = (const float*)d_in[base + 6];
    const float* s1_wr = (const float*)d_in[base + 7];
    const float* ll1_w = (const float*)d_in[base + 8];
    const float* ll1_b = (const float*)d_in[base + 9];
    const float* pp1_w = (const float*)d_in[base + 10];
    const float* pp1_b = (const float*)d_in[base + 11];
    const float* s2_wl = (const float*)d_in[base + 12];
    const float* s2_bl = (const float*)d_in[base + 13];
    const float* s2_wr = (const float*)d_in[base + 14];
    const float* ll2_w = (const float*)d_in[base + 15];
    const float* ll2_b = (const float*)d_in[base + 16];
    const float* pp2_w = (const float*)d_in[base + 17];
    const float* pp2_b = (const float*)d_in[base + 18];
    const int coff = g * 64;

    // ---- degrees & normalization ------------------------------------------
    fill_kernel<<<gblk(N, T), T, 0, stream>>>(dinv_ll, 1.0f, N);   // +1 self loop
    fill_kernel<<<gblk(N, T), T, 0, stream>>>(dinv_pp, 1.0f, N);
    fill_kernel<<<gblk(N, T), T, 0, stream>>>(cnt_lp, 0.0f, N);
    count_kernel<<<gblk(E, T), T, 0, stream>>>(ei_ll + E, dinv_ll, E);
    count_kernel<<<gblk(E, T), T, 0, stream>>>(ei_pp + E, dinv_pp, E);
    count_kernel<<<gblk(E, T), T, 0, stream>>>(ei_lp + E, cnt_lp, E);
    rsqrt_kernel<<<gblk(N, T), T, 0, stream>>>(dinv_ll, N);
    rsqrt_kernel<<<gblk(N, T), T, 0, stream>>>(dinv_pp, N);

    // ---- Layer 1: h_l = relu(GCN_ll(x_l)) ---------------------------------
    gemm_strip_wmma<128><<<gemmBlocks, T, 0, stream>>>(x_l, ll1_w, nullptr, tmp, N, 128, 0, 0, 0);
    selfloop_init_kernel<<<gblk((long long)N * 128, T), T, 0, stream>>>(tmp, dinv_ll, ll1_b, h_l, N, 128, 128, 0);
    gcn_scatter_kernel<<<gblk((long long)E * 32, T), T, 0, stream>>>(tmp, ei_ll, ei_ll + E, dinv_ll, h_l, E, 128, 128, 0);

    // ---- Layer 1: h_p = relu(SAGE_lp(x_l,x_p) + GCN_pp(x_p)) --------------
    gemm_strip_wmma<128><<<gemmBlocks, T, 0, stream>>>(x_p, pp1_w, nullptr, tmp, N, 128, 0, 0, 0);
    selfloop_init_kernel<<<gblk((long long)N * 128, T), T, 0, stream>>>(tmp, dinv_pp, pp1_b, h_p, N, 128, 128, 0);
    gcn_scatter_kernel<<<gblk((long long)E * 32, T), T, 0, stream>>>(tmp, ei_pp, ei_pp + E, dinv_pp, h_p, E, 128, 128, 0);
    // SAGE: mean(x_l)@wl == segsum(x_l@wl)/cnt  (aggregation is linear)
    gemm_strip_wmma<128><<<gemmBlocks, T, 0, stream>>>(x_l, s1_wl, nullptr, tmp, N, 128, 0, 0, 0);
    fill_kernel<<<gblk((long long)N * 128, T), T, 0, stream>>>(sbuf, 0.0f, (long long)N * 128);
    sum_scatter_kernel<<<gblk((long long)E * 32, T), T, 0, stream>>>(tmp, ei_lp, ei_lp + E, sbuf, E, 128);
    mean_add_kernel<<<gblk((long long)N * 128, T), T, 0, stream>>>(sbuf, cnt_lp, s1_bl, h_p, N, 128, 128, 0);
    // accumulate x_p @ s1_wr and fuse the ReLU (h_p finalized here)
    gemm_strip_wmma<128><<<gemmBlocks, T, 0, stream>>>(x_p, s1_wr, nullptr, h_p, N, 128, 0, 1, 1);

    relu_kernel<<<gblk((long long)N * 128, T), T, 0, stream>>>(h_l, (long long)N * 128);

    // ---- Layer 2: o_l = GCN_ll(h_l) -> d_out lnc region -------------------
    gemm_strip_wmma<64><<<gemmBlocks, T, 0, stream>>>(h_l, ll2_w, nullptr, tmp, N, 64, 0, 0, 0);
    selfloop_init_kernel<<<gblk((long long)N * 64, T), T, 0, stream>>>(tmp, dinv_ll, ll2_b, out_lnc, N, 64, 128, coff);
    gcn_scatter_kernel<<<gblk((long long)E * 16, T), T, 0, stream>>>(tmp, ei_ll, ei_ll + E, dinv_ll, out_lnc, E, 64, 128, coff);

    // ---- Layer 2: o_p = SAGE_lp(h_l,h_p) + GCN_pp(h_p) -> d_out prot ------
    gemm_strip_wmma<64><<<gemmBlocks, T, 0, stream>>>(h_p, pp2_w, nullptr, tmp, N, 64, 0, 0, 0);
    selfloop_init_kernel<<<gblk((long long)N * 64, T), T, 0, stream>>>(tmp, dinv_pp, pp2_b, out_prot, N, 64, 128, coff);
    gcn_scatter_kernel<<<gblk((long long)E * 16, T), T, 0, stream>>>(tmp, ei_pp, ei_pp + E, dinv_pp, out_prot, E, 64, 128, coff);
    // SAGE at 64 features (GEMM-first halves scatter traffic vs reference order)
    gemm_strip_wmma<64><<<gemmBlocks, T, 0, stream>>>(h_l, s2_wl, nullptr, tmp, N, 64, 0, 0, 0);
    fill_kernel<<<gblk((long long)N * 64, T), T, 0, stream>>>(sbuf, 0.0f, (long long)N * 64);
    sum_scatter_kernel<<<gblk((long long)E * 16, T), T, 0, stream>>>(tmp, ei_lp, ei_lp + E, sbuf, E, 64);
    mean_add_kernel<<<gblk((long long)N * 64, T), T, 0, stream>>>(sbuf, cnt_lp, s2_bl, out_prot, N, 64, 128, coff);
    gemm_strip_wmma<64><<<gemmBlocks, T, 0, stream>>>(h_p, s2_wr, nullptr, out_prot, N, 128, coff, 1, 0);
  }
}